// GNNLayer_83150566851123
// MI455X (gfx1250) — compile-verified
//
#include <hip/hip_runtime.h>

typedef float v2f __attribute__((ext_vector_type(2)));
typedef float v8f __attribute__((ext_vector_type(8)));

#define F       128   // input feature dim
#define NNODES  64    // nodes per graph
#define OUTD    128   // output feature dim
#define CDIM    256   // concat dim (F + OUTD)

__global__ __launch_bounds__(256) void gnn_layer_kernel(
    const float* __restrict__ x,    // (BL, 64, 128)
    const float* __restrict__ We,   // (128, 256) rows: [W1 | W2]
    const float* __restrict__ be,   // (128,)
    const float* __restrict__ Wn,   // (128, 256)
    const float* __restrict__ bn,   // (128,)
    float* __restrict__ out)        // (BL, 64, 128)
{
    __shared__ float xs [NNODES * F];     // 32 KB  x tile
    __shared__ float p1s[NNODES * OUTD];  // 32 KB  proj_i
    __shared__ float p2s[NNODES * OUTD];  // 32 KB  proj_j + be
    __shared__ float ess[NNODES * OUTD];  // 32 KB  edge_sum

    const int bl   = blockIdx.x;      // b*32 + l
    const int tid  = threadIdx.x;
    const int lane = tid & 31;
    const int wv   = tid >> 5;        // wave id 0..7
    const int l16  = lane & 15;
    const int hs   = lane >> 4;       // half-wave select (K pairing per ISA A/B layout)

    // ---- Warm L2 with the weight matrices (reused by all 128 blocks) ----
    // 256B granularity; emits global_prefetch_b8.
    {
        const int lines = (F * CDIM) / 64;             // 512 cachelines each
        for (int i = tid; i < lines; i += 256) {
            __builtin_prefetch(We + i * 64, 0, 1);
            __builtin_prefetch(Wn + i * 64, 0, 1);
        }
    }

    // ---------------- Phase 0: stage x tile into LDS ----------------
    {
        const float4* xg = (const float4*)(x + (size_t)bl * NNODES * F);
        float4* xl = (float4*)xs;
        for (int i = tid; i < NNODES * F / 4; i += 256)
            xl[i] = xg[i];
    }
    __syncthreads();

    // ------- Phase 1: P1 = x*W1^T, P2 = x*W2^T + be (WMMA f32 16x16x4) -------
    {
        const int    p  = wv >> 2;          // 0 -> W1 half, 1 -> W2 half of We rows
        const int    m0 = (wv & 3) * 16;    // node (M) tile
        const float* W  = We + p * F;       // row stride 2F, offset into W1/W2 half
        float*       Ps = p ? p2s : p1s;

        for (int nt = 0; nt < 8; ++nt) {
            const int n0   = nt * 16;
            const int brow = n0 + l16;      // output-feature row of W this lane feeds B
            // Two independent accumulator chains: halves the D->C dependency depth.
            v8f acc0 = {};
            v8f acc1 = {};
            #pragma unroll 4
            for (int k = 0; k < F; k += 8) {
                // A (16x4): lanes 0-15 hold K={k,k+1}, lanes 16-31 K={k+2,k+3}
                v2f a0 = *(const v2f*)&xs[(m0 + l16) * F + k + hs * 2];
                v2f a1 = *(const v2f*)&xs[(m0 + l16) * F + k + 4 + hs * 2];
                // B (4x16): B[kk][n] = W[n][kk]  (same half-wave K pairing)
                v2f b0 = *(const v2f*)&W[(size_t)brow * (2 * F) + k + hs * 2];
                v2f b1 = *(const v2f*)&W[(size_t)brow * (2 * F) + k + 4 + hs * 2];
                acc0 = __builtin_amdgcn_wmma_f32_16x16x4_f32(
                           false, a0, false, b0, (short)0, acc0, false, false);
                acc1 = __builtin_amdgcn_wmma_f32_16x16x4_f32(
                           false, a1, false, b1, (short)0, acc1, false, false);
            }
            v8f acc = acc0 + acc1;
            const float bias = p ? be[n0 + l16] : 0.0f;  // fold be into P2
            #pragma unroll
            for (int r = 0; r < 8; ++r) {
                const int row = m0 + r + hs * 8;         // D layout: VGPR r -> M=r / r+8
                Ps[row * OUTD + n0 + l16] = acc[r] + bias;
            }
        }
    }
    __syncthreads();

    // ------- Phase 2: edge_sum[n][o] = sum_j relu(P1[n][o] + P2b[j][o]) -------
    {
        const int o     = tid & (OUTD - 1);   // lanes -> consecutive o: no LDS conflicts
        const int nbase = (tid >> 7) * 32;    // 2 half-blocks cover n=0..63
        for (int c = 0; c < 4; ++c) {
            const int n0 = nbase + c * 8;
            float p1r[8], acc[8];
            #pragma unroll
            for (int r = 0; r < 8; ++r) {
                p1r[r] = p1s[(n0 + r) * OUTD + o];
                acc[r] = 0.0f;
            }
            for (int j = 0; j < NNODES; ++j) {
                const float pj = p2s[j * OUTD + o];   // one LDS read serves 8 rows
                #pragma unroll
                for (int r = 0; r < 8; ++r)
                    acc[r] += fmaxf(p1r[r] + pj, 0.0f);
            }
            #pragma unroll
            for (int r = 0; r < 8; ++r)
                ess[(n0 + r) * OUTD + o] = acc[r];
        }
    }
    __syncthreads();

    // ------- Phase 3: node = relu([x | edge_sum] * Wn^T + bn), K=256 -------
    {
        const int m0  = (wv & 3) * 16;
        const int nt0 = (wv >> 2) * 4;        // 8 waves x 4 N-tiles = 32 tiles
        for (int nt = nt0; nt < nt0 + 4; ++nt) {
            const int n0   = nt * 16;
            const int brow = n0 + l16;
            v8f acc0 = {};
            v8f acc1 = {};
            #pragma unroll 4
            for (int k = 0; k < CDIM; k += 8) {
                const int kk0 = k + hs * 2;       // k multiple of 8: never straddles F
                const int kk1 = k + 4 + hs * 2;
                const float* A0 = (k < F) ? &xs [(m0 + l16) * F    + kk0]
                                          : &ess[(m0 + l16) * OUTD + kk0 - F];
                const float* A1 = (k < F) ? &xs [(m0 + l16) * F    + kk1]
                                          : &ess[(m0 + l16) * OUTD + kk1 - F];
                v2f a0 = *(const v2f*)A0;
                v2f a1 = *(const v2f*)A1;
                v2f b0 = *(const v2f*)&Wn[(size_t)brow * CDIM + k + hs * 2];
                v2f b1 = *(const v2f*)&Wn[(size_t)brow * CDIM + k + 4 + hs * 2];
                acc0 = __builtin_amdgcn_wmma_f32_16x16x4_f32(
                           false, a0, false, b0, (short)0, acc0, false, false);
                acc1 = __builtin_amdgcn_wmma_f32_16x16x4_f32(
                           false, a1, false, b1, (short)0, acc1, false, false);
            }
            v8f acc = acc0 + acc1;
            const float bias = bn[n0 + l16];
            float* og = out + (size_t)bl * NNODES * OUTD;
            #pragma unroll
            for (int r = 0; r < 8; ++r) {
                const int row = m0 + r + hs * 8;
                og[row * OUTD + n0 + l16] = fmaxf(acc[r] + bias, 0.0f);
            }
        }
    }
}

extern "C" void kernel_launch(void* const* d_in, const int* in_sizes, int n_in,
                              void* d_out, int out_size, void* d_ws, size_t ws_size,
                              hipStream_t stream) {
    (void)in_sizes; (void)n_in; (void)d_ws; (void)ws_size; (void)out_size;
    const float* x  = (const float*)d_in[0];   // (4,32,64,128)
    const float* We = (const float*)d_in[1];   // (128,256)
    const float* be = (const float*)d_in[2];   // (128,)
    const float* Wn = (const float*)d_in[3];   // (128,256)
    const float* bn = (const float*)d_in[4];   // (128,)
    float* out = (float*)d_out;                // (4,32,64,128)

    dim3 grid(4 * 32);   // one workgroup per (b,l) graph
    dim3 block(256);     // 8 wave32
    gnn_layer_kernel<<<grid, block, 0, stream>>>(x, We, be, Wn, bn, out);
}